// ThresholdModel_49563922596308
// MI455X (gfx1250) — compile-verified
//
#include <hip/hip_runtime.h>
#include <hip/hip_bf16.h>
#include <math.h>

#ifndef M_PI
#define M_PI 3.14159265358979323846
#endif

// Problem geometry (fixed by the reference)
#define BATCH 16
#define CHN   3
#define IMH   512
#define IMW   512
#define NTAP  19      // 2*n+1, n = 9 for freq = 0.2
#define RAD   9

typedef __attribute__((ext_vector_type(2))) float v2f;
typedef __attribute__((ext_vector_type(8))) float v8f;

struct GaborW {
    float wv[NTAP];   // vertical factor:  exp(-0.5*d^2/sigma^2)
    float wh[NTAP];   // horizontal:       exp(-0.5*d^2/sigma^2)*cos(2*pi*f*d)/(2*pi*sigma^2)
};

// -------------------------------------------------------------------------
// Pass 1: vertical 19-tap conv with symmetric row padding, via WMMA.
// One wave computes one 16x16 tile: O(16x16) = Wv(16x36 Toeplitz) x S(36x16).
// 9 x V_WMMA_F32_16X16X4_F32, K-chunks of 4.
//
// f32 WMMA operand layouts (CDNA5 ISA 7.12.2):
//   A 16x4:  lane L (h=L>>4, m=L&15): a[j] = A[m, 2h + j]     (chunk-local K)
//   B 4x16:  lane L:                  b[j] = B[2h + j, L&15]
//   C/D:     vgpr g, lane L:          D[g + 8h, L&15]
// -------------------------------------------------------------------------
#define WPAD 56   // zero-padded weight table: index (k - m) + 16 in [1, 51]

__global__ __launch_bounds__(128)
void gabor_vpass_wmma(const float* __restrict__ x, float* __restrict__ vtmp, GaborW wts)
{
    __shared__ float wpad[WPAD];
    const int tid = threadIdx.x;
    for (int i = tid; i < WPAD; i += 128) wpad[i] = 0.0f;
    __syncthreads();
    if (tid < NTAP) wpad[16 + tid] = wts.wv[tid];
    __syncthreads();

    const int lane = tid & 31;
    const int wave = tid >> 5;                     // 4 waves / block, 1 tile each
    const int tx   = (blockIdx.x * 4 + wave) * 16; // tile col origin
    const int ty   = blockIdx.y * 16;              // tile row origin
    const int b    = blockIdx.z;

    const float* __restrict__ plane = x + (size_t)b * CHN * IMH * IMW; // channel 0
    float* __restrict__ outp        = vtmp + (size_t)b * IMH * IMW;

    const int m    = lane & 15;       // output row within tile / B column
    const int half = lane >> 4;
    const int col  = tx + m;          // B column -> global image column

    // ---- A operand: Toeplitz weights Wv[m, k] = wv[k-m], branchless via
    //      zero-padded LDS table; hoisted fully into registers. ----
    float av[18];
    #pragma unroll
    for (int kk = 0; kk < 9; ++kk) {
        #pragma unroll
        for (int j = 0; j < 2; ++j) {
            const int k = kk * 4 + half * 2 + j;           // K in [0,36)
            av[kk * 2 + j] = wpad[16 + k - m];             // index in [1,51]
        }
    }

    // ---- B operand: 18 input rows per lane, preloaded before the WMMA
    //      chain so the loads overlap. Interior tiles (block-uniform test)
    //      use one base address + immediate offsets; edge tiles reflect. ----
    float bv[18];
    if (ty >= 16 && ty <= IMH - 32) {                      // rows ty-9..ty+26 all in-range
        const float* __restrict__ base =
            plane + (size_t)(ty - RAD + half * 2) * IMW + col;
        #pragma unroll
        for (int kk = 0; kk < 9; ++kk) {
            #pragma unroll
            for (int j = 0; j < 2; ++j)
                bv[kk * 2 + j] = base[(size_t)(kk * 4 + j) * IMW];
        }
    } else {
        #pragma unroll
        for (int kk = 0; kk < 9; ++kk) {
            #pragma unroll
            for (int j = 0; j < 2; ++j) {
                const int k = kk * 4 + half * 2 + j;
                int r = ty + k - RAD;                      // in [-9, 522]
                r = (r < 0)       ? (-1 - r)          : r; // symmetric reflect
                r = (r > IMH - 1) ? (2 * IMH - 1 - r) : r;
                bv[kk * 2 + j] = plane[(size_t)r * IMW + col];
            }
        }
    }

    // ---- 9-step WMMA accumulation chain ----
    v8f acc = {};
    #pragma unroll
    for (int kk = 0; kk < 9; ++kk) {
        v2f a, bb;
        a[0]  = av[kk * 2];  a[1]  = av[kk * 2 + 1];
        bb[0] = bv[kk * 2];  bb[1] = bv[kk * 2 + 1];
        acc = __builtin_amdgcn_wmma_f32_16x16x4_f32(
                  /*neg_a=*/false, a, /*neg_b=*/false, bb,
                  /*c_mod=*/(short)0, acc, /*reuse_a=*/false, /*reuse_b=*/false);
    }

    #pragma unroll
    for (int g = 0; g < 8; ++g) {
        const int row = ty + g + half * 8;                 // D layout
        outp[(size_t)row * IMW + col] = acc[g];
    }
}

// -------------------------------------------------------------------------
// Pass 2: horizontal 19-tap conv (symmetric col padding) + threshold(>0.5)
//         + cross dilation (zero-padded halo), fused per 16x64 tile.
// -------------------------------------------------------------------------
#define T2H 16
#define T2W 64
#define VLD (T2W + 2 + 2 * RAD)   // 84 staged vtmp columns
#define MKW (T2W + 2)             // 66 mask columns (dilation halo)

__global__ __launch_bounds__(256)
void gabor_hpass_dilate(const float* __restrict__ vtmp, float* __restrict__ out, GaborW wts)
{
    __shared__ float whs[NTAP];
    __shared__ float vt[T2H + 2][VLD];
    __shared__ float mk[T2H + 2][MKW];

    const int tid = threadIdx.x;
    if (tid < NTAP) whs[tid] = wts.wh[tid];

    const int b  = blockIdx.z;
    const int y0 = blockIdx.y * T2H;
    const int x0 = blockIdx.x * T2W;
    const float* __restrict__ plane = vtmp + (size_t)b * IMH * IMW;

    // Stage vtmp rows y0-1..y0+16 (clamped; out-of-image rows only feed mask
    // entries that are forced to 0), cols x0-10..x0+73 (symmetric reflect).
    for (int idx = tid; idx < (T2H + 2) * VLD; idx += 256) {
        const int rr = idx / VLD, cc = idx % VLD;
        int r = y0 - 1 + rr;
        r = (r < 0) ? 0 : (r > IMH - 1 ? IMH - 1 : r);
        int c = x0 - 10 + cc;
        c = (c < 0)       ? (-1 - c)          : c;
        c = (c > IMW - 1) ? (2 * IMW - 1 - c) : c;
        vt[rr][cc] = plane[(size_t)r * IMW + c];
    }
    __syncthreads();

    // Mask (conv > 0.5) on the haloed 18x66 region; zero outside the image
    // (matches the reference's zero-padding before dilation).
    for (int idx = tid; idx < (T2H + 2) * MKW; idx += 256) {
        const int rr = idx / MKW, cc = idx % MKW;
        const int r = y0 - 1 + rr;
        const int c = x0 - 1 + cc;
        float mval = 0.0f;
        if (r >= 0 && r < IMH && c >= 0 && c < IMW) {
            float s = 0.0f;
            #pragma unroll
            for (int t = 0; t < NTAP; ++t)
                s = fmaf(whs[t], vt[rr][cc + t], s);
            mval = (s > 0.5f) ? 1.0f : 0.0f;
        }
        mk[rr][cc] = mval;
    }
    __syncthreads();

    float* __restrict__ op = out + (size_t)b * IMH * IMW;
    for (int idx = tid; idx < T2H * T2W; idx += 256) {
        const int rr = idx / T2W, cc = idx % T2W;
        float v = mk[rr + 1][cc + 1];
        v = fmaxf(v, mk[rr    ][cc + 1]);
        v = fmaxf(v, mk[rr + 2][cc + 1]);
        v = fmaxf(v, mk[rr + 1][cc    ]);
        v = fmaxf(v, mk[rr + 1][cc + 2]);
        op[(size_t)(y0 + rr) * IMW + (x0 + cc)] = v;
    }
}

// -------------------------------------------------------------------------
// Host side: double-precision weights (matches numpy f64 kernel build),
// passed by value as kernargs.
// -------------------------------------------------------------------------
static GaborW make_weights()
{
    GaborW w;
    const double freq      = 0.2;
    const double prefactor = 1.0 / M_PI * sqrt(log(2.0) / 2.0) * 3.0;
    const double sigma     = prefactor / freq;
    const double s2        = sigma * sigma;
    const double norm      = 1.0 / (2.0 * M_PI * s2);
    for (int t = 0; t < NTAP; ++t) {
        const double d = (double)(t - RAD);
        const double e = exp(-0.5 * d * d / s2);
        w.wv[t] = (float)e;
        w.wh[t] = (float)(e * cos(2.0 * M_PI * freq * d) * norm);
    }
    return w;
}

extern "C" void kernel_launch(void* const* d_in, const int* in_sizes, int n_in,
                              void* d_out, int out_size, void* d_ws, size_t ws_size,
                              hipStream_t stream)
{
    const float* x  = (const float*)d_in[0];          // (16,3,512,512) f32
    float* out      = (float*)d_out;                  // (16,1,512,512) f32
    float* vtmp     = (float*)d_ws;                   // 16*512*512 f32 = 16 MiB

    const GaborW w = make_weights();

    // Pass 1: WMMA vertical conv. 4 waves/block, 1 tile/wave.
    dim3 g1(IMW / (16 * 4), IMH / 16, BATCH);         // (8, 32, 16)
    gabor_vpass_wmma<<<g1, 128, 0, stream>>>(x, vtmp, w);

    // Pass 2: horizontal conv + threshold + dilation, fused.
    dim3 g2(IMW / T2W, IMH / T2H, BATCH);             // (8, 32, 16)
    gabor_hpass_dilate<<<g2, 256, 0, stream>>>(vtmp, out, w);
}